// DropConnect_57432302682892
// MI455X (gfx1250) — compile-verified
//
#include <hip/hip_runtime.h>

typedef float v2f __attribute__((ext_vector_type(2)));
typedef float v8f __attribute__((ext_vector_type(8)));

#define B_ 64
#define D_ 1024
#define O_ 2048

// Each wave computes Z[b, obase:obase+16] = sum_d X[b,d] * sign(W[d,o]) * Werr[b,d,o]
// via V_WMMA_F32_16X16X4_F32 with A = broadcast x-chunk (all 16 rows identical),
// B = copysign(Werr, W) (exactly {-1,-0,+1} since Werr in {0.0,1.0}).
__global__ __launch_bounds__(128) void dropconnect_wmma_kernel(
    const float* __restrict__ X, const float* __restrict__ W,
    const float* __restrict__ bias, const float* __restrict__ Werr,
    const float* __restrict__ Berr, float* __restrict__ out)
{
    __shared__ float xs[D_];

    const int tid = threadIdx.x;
    const int b = blockIdx.x >> 5;                 // 32 o-blocks per batch sample
    const int obase_block = (blockIdx.x & 31) << 6; // 64 outputs per block

    // Cooperative, divergence-free stage of this sample's X row into LDS.
    const float* xrow = X + (size_t)b * D_;
    #pragma unroll
    for (int i = 0; i < D_ / 128; ++i)
        xs[tid + i * 128] = xrow[tid + i * 128];
    __syncthreads();

    const int wave  = tid >> 5;
    const int lane  = tid & 31;
    const int n     = lane & 15;   // output column within the 16-wide tile
    const int khalf = lane >> 4;   // 0: K=0,1   1: K=2,3  (A & B operand halves)
    const int obase = obase_block + (wave << 4);
    const int o     = obase + n;

    // B-operand element (k, n) lives at (d = dbase + 2*khalf + {0,1}, o = obase+n).
    const float* werr_p = Werr + ((size_t)b * D_ + (size_t)(khalf << 1)) * O_ + o;
    const float* w_p    = W    + (size_t)(khalf << 1) * O_ + o;
    const float* xp     = xs + (khalf << 1);

    v8f acc = {0.f, 0.f, 0.f, 0.f, 0.f, 0.f, 0.f, 0.f};

    #pragma unroll 4
    for (int it = 0; it < D_ / 4; ++it) {
        // Werr streams through once: non-temporal so it doesn't thrash L2;
        // W (8 MB) stays cacheable and is reused by all 64 samples.
        float m0 = __builtin_nontemporal_load(werr_p);
        float m1 = __builtin_nontemporal_load(werr_p + O_);
        float w0 = w_p[0];
        float w1 = w_p[O_];

        v2f bb;
        bb.x = __builtin_copysignf(m0, w0);  // sign(W) * Werr, exact in f32
        bb.y = __builtin_copysignf(m1, w1);

        v2f aa;                               // A row = x[dbase..dbase+3], all rows equal
        aa.x = xp[0];
        aa.y = xp[1];

        acc = __builtin_amdgcn_wmma_f32_16x16x4_f32(
            /*neg_a=*/false, aa, /*neg_b=*/false, bb,
            /*c_mod=*/(short)0, acc, /*reuse_a=*/false, /*reuse_b=*/false);

        werr_p += 4 * O_;
        w_p    += 4 * O_;
        xp     += 4;
    }

    // All 16 rows of D are identical; row M=0 sits in acc[0] for lanes 0..15.
    if (lane < 16) {
        size_t idx = (size_t)b * O_ + o;
        out[idx] = acc[0] + bias[o] * Berr[idx];
    }
}

extern "C" void kernel_launch(void* const* d_in, const int* in_sizes, int n_in,
                              void* d_out, int out_size, void* d_ws, size_t ws_size,
                              hipStream_t stream) {
    const float* X    = (const float*)d_in[0];
    const float* W    = (const float*)d_in[1];
    const float* bias = (const float*)d_in[2];
    const float* Werr = (const float*)d_in[3];
    const float* Berr = (const float*)d_in[4];
    float* out = (float*)d_out;

    dim3 grid(B_ * (O_ / 64));   // 2048 blocks, 4 waves each: one (b, 16-o tile) per wave
    dropconnect_wmma_kernel<<<grid, 128, 0, stream>>>(X, W, bias, Werr, Berr, out);
}